// GroupTokenizer_20040317403184
// MI455X (gfx1250) — compile-verified
//
#include <hip/hip_runtime.h>
#include <stdint.h>

#define CC 8
#define KK 64
#define BLOCK 256

typedef float    v4f __attribute__((ext_vector_type(4)));
typedef uint32_t v4u __attribute__((ext_vector_type(4)));
typedef int      v8i __attribute__((ext_vector_type(8)));
typedef int      v4i __attribute__((ext_vector_type(4)));

__global__ __launch_bounds__(BLOCK)
void group_tokenizer_kernel(const float* __restrict__ y,
                            const float* __restrict__ ledges,
                            const float* __restrict__ redges,
                            int*   __restrict__ labels_out,
                            float* __restrict__ reg_out,
                            int n)
{
    __shared__ float raw[2 * CC * KK];   // staged [C][K]: left then right
    __shared__ float lt[KK * CC];        // transposed [K][C]
    __shared__ float rt[KK * CC];
    __shared__ int   lab_s[BLOCK];
    __shared__ float del_s[BLOCK];

    const int tid = threadIdx.x;

    // ---- Stage edges into LDS via the Tensor Data Mover (CDNA5 TDM) ----
#if __has_builtin(__builtin_amdgcn_tensor_load_to_lds)
    if (tid < 32) {                      // one wave issues the DMA
        const uint32_t lds_base = (uint32_t)(uintptr_t)(&raw[0]);
        const uint64_t gl = (uint64_t)(uintptr_t)ledges;
        const uint64_t gr = (uint64_t)(uintptr_t)redges;

        // D# group1: data_size=4B(code 2), tensor 64x8, tile 64x8, dim0 stride 64
        v8i g1 = {
            /*w0*/ (2 << 16),                 // workgroup_mask=0 | data_size=2
            /*w1*/ (KK & 0xffff) << 16,       // atomic_barrier=0 | tensor_dim0 lo16
            /*w2*/ (CC & 0xffff) << 16,       // tensor_dim0 hi16(0) | tensor_dim1 lo16
            /*w3*/ (KK & 0xffff) << 16,       // tensor_dim1 hi16(0) | tile_dim0
            /*w4*/ CC,                        // tile_dim1=8 | tile_dim2=0
            /*w5*/ KK,                        // tensor_dim0_stride lo32
            /*w6*/ 0,                         // stride0 hi16 | stride1 lo16 (unused 2D)
            /*w7*/ 0
        };
        v4i z4 = {0, 0, 0, 0};
        v8i z8 = {0, 0, 0, 0, 0, 0, 0, 0};

        v4u g0l, g0r;
        g0l[0] = 1u;                                  // count=1, is_restore=0
        g0l[1] = lds_base;                            // LDS byte address
        g0l[2] = (uint32_t)gl;                        // global_addr lo32
        g0l[3] = (uint32_t)((gl >> 32) & 0x01ffffffu) | 0x80000000u; // hi25 | type=2
        g0r[0] = 1u;
        g0r[1] = lds_base + CC * KK * 4;
        g0r[2] = (uint32_t)gr;
        g0r[3] = (uint32_t)((gr >> 32) & 0x01ffffffu) | 0x80000000u;

        __builtin_amdgcn_tensor_load_to_lds(g0l, g1, z4, z4, z8, 0);
        __builtin_amdgcn_tensor_load_to_lds(g0r, g1, z4, z4, z8, 0);
        __builtin_amdgcn_s_wait_tensorcnt(0);
    }
#else
    for (int i = tid; i < CC * KK; i += BLOCK) {
        raw[i]           = ledges[i];
        raw[CC * KK + i] = redges[i];
    }
#endif
    __syncthreads();

    // Transpose [C][K] -> [K][C] so the scan is LDS-bank-conflict-free (wave32)
    for (int i = tid; i < CC * KK; i += BLOCK) {
        const int c = i >> 6, k = i & 63;
        lt[k * CC + c] = raw[i];
        rt[k * CC + c] = raw[CC * KK + i];
    }
    __syncthreads();

    // ---- Per-element bin search (first-match semantics, like argmax) ----
    const int   elem = blockIdx.x * BLOCK + tid;
    const bool  ok   = (elem < n);
    const float yv   = ok ? y[elem] : 0.0f;
    const int   c    = elem & (CC - 1);

    int   label = -1;
    float ls = 0.0f, rs = 0.0f;
#pragma unroll 8
    for (int k = 0; k < KK; ++k) {
        const float l  = lt[k * CC + c];
        const float r  = rt[k * CC + c];
        const bool  in = (yv >= l) && (yv < r);
        if (label < 0) {
            if (in)            { label = k;      ls = l; rs = r; }
            else if (k == KK-1){ label = KK - 1; ls = l; rs = r; }
        }
    }
    const float width = fmaxf(rs - ls, 1e-12f);
    float delta = (yv - ls) / width;
    delta = fminf(fmaxf(delta, 0.0f), 1.0f);

    if (ok) labels_out[elem] = label;
    lab_s[tid] = label;
    del_s[tid] = delta;
    __syncthreads();

    // ---- Coalesced NT float4 stores of reg: 256 elems * 64 k = 4096 x b128 ----
    v4f* regb = (v4f*)(reg_out + (size_t)blockIdx.x * BLOCK * KK);
#pragma unroll
    for (int it = 0; it < 16; ++it) {
        const int i  = it * BLOCK + tid;   // float4 index within block tile
        const int e  = i >> 4;             // element within block
        const int kb = (i & 15) << 2;      // k base of this float4
        const int   lbl = lab_s[e];
        const float dv  = del_s[e];
        v4f v;
        v.x = (lbl == kb + 0) ? dv : -1.0f;
        v.y = (lbl == kb + 1) ? dv : -1.0f;
        v.z = (lbl == kb + 2) ? dv : -1.0f;
        v.w = (lbl == kb + 3) ? dv : -1.0f;
        if (blockIdx.x * BLOCK + e < n)
            __builtin_nontemporal_store(v, &regb[i]);
    }
}

extern "C" void kernel_launch(void* const* d_in, const int* in_sizes, int n_in,
                              void* d_out, int out_size, void* d_ws, size_t ws_size,
                              hipStream_t stream) {
    const float* y = (const float*)d_in[0];
    const float* l = (const float*)d_in[1];
    const float* r = (const float*)d_in[2];
    const int n = in_sizes[0];                 // B*T*C = 1,048,576

    int*   labels = (int*)d_out;               // first N int32 labels
    float* reg    = (float*)d_out + n;         // then N*K f32 reg

    const int blocks = (n + BLOCK - 1) / BLOCK;
    group_tokenizer_kernel<<<dim3(blocks), dim3(BLOCK), 0, stream>>>(
        y, l, r, labels, reg, n);
}